// LSQGradient_395136991944
// MI455X (gfx1250) — compile-verified
//
#include <hip/hip_runtime.h>
#include <hip/hip_bf16.h>

#define LSQ_EPS 1e-8f

typedef float v8f __attribute__((ext_vector_type(8)));
typedef float v2f __attribute__((ext_vector_type(2)));

// ---------------------------------------------------------------------------
// relaxed agent-scope fp32 atomic add (should lower to global_atomic_add_f32,
// non-returning since result is unused)
__device__ __forceinline__ void atomAddF32(float* p, float v) {
    __hip_atomic_fetch_add(p, v, __ATOMIC_RELAXED, __HIP_MEMORY_SCOPE_AGENT);
}

// ---------------------------------------------------------------------------
__global__ void lsq_zero_kernel(float* __restrict__ acc, int n) {
    int i = blockIdx.x * blockDim.x + threadIdx.x;
    int stride = gridDim.x * blockDim.x;
    for (; i < n; i += stride) acc[i] = 0.0f;
}

// ---------------------------------------------------------------------------
// per-edge contribution: acc[node] layout = [a00,a01,a02,a11,a12,a22,b0,b1,b2]
__device__ __forceinline__ void lsq_edge(const float* __restrict__ pos,
                                         const float* __restrict__ phi,
                                         float* __restrict__ acc,
                                         int r, int c) {
    float prx = pos[3 * (size_t)r + 0];
    float pry = pos[3 * (size_t)r + 1];
    float prz = pos[3 * (size_t)r + 2];
    float pcx = pos[3 * (size_t)c + 0];
    float pcy = pos[3 * (size_t)c + 1];
    float pcz = pos[3 * (size_t)c + 2];
    float dx = pcx - prx, dy = pcy - pry, dz = pcz - prz;
    float d2 = dx * dx + dy * dy + dz * dz;
    float s  = sqrtf(d2) + LSQ_EPS;
    float w  = 1.0f / (s * s);
    float dphi = phi[c] - phi[r];
    float wd = w * dphi;
    float* a = acc + (size_t)r * 9;
    atomAddF32(a + 0, w * dx * dx);
    atomAddF32(a + 1, w * dx * dy);
    atomAddF32(a + 2, w * dx * dz);
    atomAddF32(a + 3, w * dy * dy);
    atomAddF32(a + 4, w * dy * dz);
    atomAddF32(a + 5, w * dz * dz);
    atomAddF32(a + 6, wd * dx);
    atomAddF32(a + 7, wd * dy);
    atomAddF32(a + 8, wd * dz);
}

// 4 edges / thread via int4 loads of both index streams (E % 4 == 0 path)
__global__ void lsq_edge_kernel_vec4(const float* __restrict__ pos,
                                     const float* __restrict__ phi,
                                     const int* __restrict__ rows,
                                     const int* __restrict__ cols,
                                     float* __restrict__ acc, int e4) {
    const int4* __restrict__ r4p = (const int4*)rows;
    const int4* __restrict__ c4p = (const int4*)cols;
    int i = blockIdx.x * blockDim.x + threadIdx.x;
    int stride = gridDim.x * blockDim.x;
    for (; i < e4; i += stride) {
        // run-ahead prefetch of the 51MB index stream (global_prefetch_b8)
        __builtin_prefetch((const void*)(r4p + i + stride), 0, 1);
        __builtin_prefetch((const void*)(c4p + i + stride), 0, 1);
        int4 r4 = r4p[i];
        int4 c4 = c4p[i];
        lsq_edge(pos, phi, acc, r4.x, c4.x);
        lsq_edge(pos, phi, acc, r4.y, c4.y);
        lsq_edge(pos, phi, acc, r4.z, c4.z);
        lsq_edge(pos, phi, acc, r4.w, c4.w);
    }
}

__global__ void lsq_edge_kernel_scalar(const float* __restrict__ pos,
                                       const float* __restrict__ phi,
                                       const int* __restrict__ rows,
                                       const int* __restrict__ cols,
                                       float* __restrict__ acc, int e) {
    int i = blockIdx.x * blockDim.x + threadIdx.x;
    int stride = gridDim.x * blockDim.x;
    for (; i < e; i += stride) lsq_edge(pos, phi, acc, rows[i], cols[i]);
}

// ---------------------------------------------------------------------------
// Solve (A + eps I) g = b per node via symmetric Cramer inverse, then the
// batched 3x3 matvec g = Ainv*b done on V_WMMA_F32_16X16X4_F32.
//
// Mapping (one wave = 16 nodes): lanes 0-15 and 16-31 both compute node
// base + (lane&15), so per ISA 7.12.2:
//   A (16x4, 2 VGPRs): lane<16 -> {Ainv[c][0], Ainv[c][1]},
//                      lane>=16 -> {Ainv[c][2], 0}           (all lane-local)
//   B (4x16, 2 VGPRs): lane<16 -> {b0, b1}, lane>=16 -> {b2, 0}
// Diagonal D[i][i] = Ainv[c]·b of node i; it lives in lane i (i<8, elem i)
// and lane i+16 (i>=8, elem i-8), i.e. elem = lane&7.
__global__ void lsq_solve_kernel(const float* __restrict__ acc,
                                 float* __restrict__ out, int N) {
    int tid  = blockIdx.x * blockDim.x + threadIdx.x;
    int lane = threadIdx.x & 31;
    int wave = tid >> 5;
    int base = wave * 16;
    int node = base + (lane & 15);
    int nc   = node < N ? node : (N - 1);

    const float* a = acc + (size_t)nc * 9;
    float m00 = a[0] + LSQ_EPS;
    float m01 = a[1];
    float m02 = a[2];
    float m11 = a[3] + LSQ_EPS;
    float m12 = a[4];
    float m22 = a[5] + LSQ_EPS;
    float b0 = a[6], b1 = a[7], b2 = a[8];

    // symmetric adjugate
    float i00 = m11 * m22 - m12 * m12;
    float i01 = m02 * m12 - m01 * m22;
    float i02 = m01 * m12 - m02 * m11;
    float i11 = m00 * m22 - m02 * m02;
    float i12 = m01 * m02 - m00 * m12;
    float i22 = m00 * m11 - m01 * m01;
    float det  = m00 * i00 + m01 * i01 + m02 * i02;
    float rdet = 1.0f / det;
    i00 *= rdet; i01 *= rdet; i02 *= rdet;
    i11 *= rdet; i12 *= rdet; i22 *= rdet;

    bool lo = lane < 16;
    v2f vb;
    vb.x = lo ? b0 : b2;
    vb.y = lo ? b1 : 0.0f;

    bool diagActive = (lane < 8) || (lane >= 24);
    int  dnode = base + (lo ? lane : lane - 16);

#define SOLVE_COMP(C, R0, R1, R2)                                             \
    do {                                                                      \
        v2f va;                                                               \
        va.x = lo ? (R0) : (R2);                                              \
        va.y = lo ? (R1) : 0.0f;                                              \
        v8f dd = {};                                                          \
        dd = __builtin_amdgcn_wmma_f32_16x16x4_f32(                           \
            false, va, false, vb, (short)0, dd, false, false);                \
        int  e   = lane & 7;                                                  \
        float val = dd[0];                                                    \
        val = (e == 1) ? dd[1] : val;                                         \
        val = (e == 2) ? dd[2] : val;                                         \
        val = (e == 3) ? dd[3] : val;                                         \
        val = (e == 4) ? dd[4] : val;                                         \
        val = (e == 5) ? dd[5] : val;                                         \
        val = (e == 6) ? dd[6] : val;                                         \
        val = (e == 7) ? dd[7] : val;                                         \
        if (diagActive && dnode < N) out[(size_t)dnode * 3 + (C)] = val;      \
    } while (0)

    SOLVE_COMP(0, i00, i01, i02);
    SOLVE_COMP(1, i01, i11, i12);
    SOLVE_COMP(2, i02, i12, i22);
#undef SOLVE_COMP
}

// ---------------------------------------------------------------------------
extern "C" void kernel_launch(void* const* d_in, const int* in_sizes, int n_in,
                              void* d_out, int out_size, void* d_ws, size_t ws_size,
                              hipStream_t stream) {
    const float* pos = (const float*)d_in[0];
    const float* phi = (const float*)d_in[1];
    const int*   ei  = (const int*)d_in[2];
    float*       out = (float*)d_out;
    float*       acc = (float*)d_ws;

    const int N = in_sizes[1];            // phi has one entry per node
    const int E = in_sizes[2] / 2;        // edge_index is (2, E) flat

    const int*   rows = ei;
    const int*   cols = ei + (size_t)E;

    // 1) zero accumulators (9 floats per node)
    {
        int n9 = N * 9;
        int blocks = (n9 + 255) / 256;
        if (blocks > 2048) blocks = 2048;
        lsq_zero_kernel<<<blocks, 256, 0, stream>>>(acc, n9);
    }

    // 2) edge scatter
    if ((E & 3) == 0) {
        int e4 = E >> 2;
        int blocks = (e4 + 255) / 256;
        if (blocks > 16384) blocks = 16384;
        lsq_edge_kernel_vec4<<<blocks, 256, 0, stream>>>(pos, phi, rows, cols,
                                                         acc, e4);
    } else {
        int blocks = (E + 255) / 256;
        if (blocks > 16384) blocks = 16384;
        lsq_edge_kernel_scalar<<<blocks, 256, 0, stream>>>(pos, phi, rows, cols,
                                                           acc, E);
    }

    // 3) per-node solve + WMMA batched matvec
    {
        int waves   = (N + 15) / 16;
        int threads = waves * 32;
        int blocks  = (threads + 255) / 256;
        lsq_solve_kernel<<<blocks, 256, 0, stream>>>(acc, out, N);
    }
}